// BilinearAttention_11287174054065
// MI455X (gfx1250) — compile-verified
//
#include <hip/hip_runtime.h>
#include <hip/hip_bf16.h>

typedef __attribute__((ext_vector_type(2))) float v2f;
typedef __attribute__((ext_vector_type(8))) float v8f;

#define B_  8
#define Q_  256
#define KK_ 512
#define QS_ 512
#define KS_ 512
#define H_  256
#define VD_ 512
#define QB_ 4      // q-rows per score block (raises tanh/byte 4x)

__device__ __forceinline__ float fast_tanh(float x) {
#if __has_builtin(__builtin_amdgcn_tanhf)
  return __builtin_amdgcn_tanhf(x);          // native v_tanh_f32
#else
  float r;
  asm("v_tanh_f32 %0, %1" : "=v"(r) : "v"(x));   // force native TRANS op
  return r;
#endif
}

__device__ __forceinline__ v8f wmma_f32_4(v2f a, v2f b, v8f c) {
  // V_WMMA_F32_16X16X4_F32 : D = A(16x4) * B(4x16) + C(16x16), fp32
  return __builtin_amdgcn_wmma_f32_16x16x4_f32(false, a, false, b, (short)0, c,
                                               false, false);
}

// ---------------------------------------------------------------------------
// Kernel 1: WQ[b,q,h] = sum_e query[b,q,e] * Wq[e,h] + bq[h]   (fp32 WMMA)
// ---------------------------------------------------------------------------
__global__ __launch_bounds__(32)
void proj_q_kernel(const float* __restrict__ X, const float* __restrict__ W,
                   const float* __restrict__ bias, float* __restrict__ Out) {
  const int lane = threadIdx.x;
  const int lr = lane & 15, half = lane >> 4;
  const int n0 = blockIdx.x * 16;   // H tile
  const int m0 = blockIdx.y * 16;   // Q tile
  const int b  = blockIdx.z;

  const float* Xb = X + (size_t)b * Q_ * QS_;
  v8f acc = {};
  for (int e0 = 0; e0 < QS_; e0 += 4) {
    const float* arow = Xb + (size_t)(m0 + lr) * QS_ + (e0 + 2 * half);
    v2f a;  a.x = arow[0];  a.y = arow[1];
    v2f bm;
    bm.x = W[(size_t)(e0 + 2 * half)     * H_ + (n0 + lr)];
    bm.y = W[(size_t)(e0 + 2 * half + 1) * H_ + (n0 + lr)];
    acc = wmma_f32_4(a, bm, acc);
  }
  const float bv = bias[n0 + lr];
  float* outb = Out + (size_t)b * Q_ * H_;
#pragma unroll
  for (int r = 0; r < 8; ++r)
    outb[(size_t)(m0 + r + 8 * half) * H_ + (n0 + lr)] = acc[r] + bv;
}

// ---------------------------------------------------------------------------
// Kernel 2: UHt[b,h,k] = sum_e Wk[e,h] * key[b,k,e]   (fp32 WMMA; output is
// stored (B,H,K)-transposed so score-kernel lanes read contiguous k)
// ---------------------------------------------------------------------------
__global__ __launch_bounds__(32)
void proj_k_kernel(const float* __restrict__ Key, const float* __restrict__ Wk,
                   float* __restrict__ UHt) {
  const int lane = threadIdx.x;
  const int lr = lane & 15, half = lane >> 4;
  const int n0 = blockIdx.x * 16;   // K tile
  const int m0 = blockIdx.y * 16;   // H tile
  const int b  = blockIdx.z;

  const float* Kb = Key + (size_t)b * KK_ * KS_;
  v8f acc = {};
  for (int e0 = 0; e0 < KS_; e0 += 4) {
    v2f a;
    a.x = Wk[(size_t)(e0 + 2 * half)     * H_ + (m0 + lr)];
    a.y = Wk[(size_t)(e0 + 2 * half + 1) * H_ + (m0 + lr)];
    const float* brow = Kb + (size_t)(n0 + lr) * KS_ + (e0 + 2 * half);
    v2f bm;  bm.x = brow[0];  bm.y = brow[1];
    acc = wmma_f32_4(a, bm, acc);
  }
  float* ub = UHt + (size_t)b * H_ * KK_;
#pragma unroll
  for (int r = 0; r < 8; ++r)
    ub[(size_t)(m0 + r + 8 * half) * KK_ + (n0 + lr)] = acc[r];
}

// ---------------------------------------------------------------------------
// Kernel 3: fused scores + softmax, QB_=4 q-rows per block.
// Each u-value loaded once feeds 4 wq rows -> 8 v_tanh_f32 per 8B of L2
// traffic; kernel is TRANS-unit bound (the hardware floor for this op).
// ---------------------------------------------------------------------------
__global__ __launch_bounds__(256)
void scores_softmax_kernel(const float* __restrict__ WQp,
                           const float* __restrict__ UHt,
                           const float* __restrict__ vvec,
                           float* __restrict__ AttnOut) {
  __shared__ float red[256];
  __shared__ float sw[QB_ * H_];
  __shared__ float sv[H_];
  const int t = threadIdx.x;
  const int q0 = blockIdx.x * QB_;
  const int b  = blockIdx.y;

  // stage the 4 wq rows and the v vector in LDS
#pragma unroll
  for (int i = t; i < QB_ * H_; i += 256) {
    const int g = i >> 8;            // H_ == 256
    const int h = i & (H_ - 1);
    sw[i] = WQp[((size_t)b * Q_ + q0 + g) * H_ + h];
  }
  sv[t] = vvec[t];
  __syncthreads();

  const float* uh = UHt + (size_t)b * H_ * KK_;
  float a0[QB_], a1[QB_];
#pragma unroll
  for (int g = 0; g < QB_; ++g) { a0[g] = 0.f; a1[g] = 0.f; }

#pragma unroll 2
  for (int h = 0; h < H_; ++h) {
    const float vh = sv[h];
    const float u0 = uh[(size_t)h * KK_ + t];
    const float u1 = uh[(size_t)h * KK_ + t + 256];
#pragma unroll
    for (int g = 0; g < QB_; ++g) {
      const float w = sw[g * H_ + h];
      a0[g] = __builtin_fmaf(vh, fast_tanh(w + u0), a0[g]);
      a1[g] = __builtin_fmaf(vh, fast_tanh(w + u1), a1[g]);
    }
  }

  // per-row softmax (rows processed sequentially; reductions are cheap)
  for (int g = 0; g < QB_; ++g) {
    red[t] = fmaxf(a0[g], a1[g]);
    __syncthreads();
    for (int s = 128; s > 0; s >>= 1) {
      if (t < s) red[t] = fmaxf(red[t], red[t + s]);
      __syncthreads();
    }
    const float rowmax = red[0];
    __syncthreads();

    const float e0 = __expf(a0[g] - rowmax);
    const float e1 = __expf(a1[g] - rowmax);
    red[t] = e0 + e1;
    __syncthreads();
    for (int s = 128; s > 0; s >>= 1) {
      if (t < s) red[t] += red[t + s];
      __syncthreads();
    }
    const float inv = 1.0f / red[0];
    __syncthreads();                 // red reused next g

    float* arow = AttnOut + ((size_t)b * Q_ + q0 + g) * KK_;
    arow[t]       = e0 * inv;
    arow[t + 256] = e1 * inv;
  }
}

// ---------------------------------------------------------------------------
// Kernel 4: out[b,q,d] = sum_k attn[b,q,k] * value[b,k,d]   (fp32 WMMA)
// ---------------------------------------------------------------------------
__global__ __launch_bounds__(32)
void av_kernel(const float* __restrict__ Attn, const float* __restrict__ V,
               float* __restrict__ Out) {
  const int lane = threadIdx.x;
  const int lr = lane & 15, half = lane >> 4;
  const int n0 = blockIdx.x * 16;   // VD tile
  const int m0 = blockIdx.y * 16;   // Q tile
  const int b  = blockIdx.z;

  const float* Ab = Attn + (size_t)b * Q_ * KK_;
  const float* Vb = V    + (size_t)b * KK_ * VD_;
  v8f acc = {};
  for (int k0 = 0; k0 < KK_; k0 += 4) {
    const float* arow = Ab + (size_t)(m0 + lr) * KK_ + (k0 + 2 * half);
    v2f a;  a.x = arow[0];  a.y = arow[1];
    v2f bm;
    bm.x = Vb[(size_t)(k0 + 2 * half)     * VD_ + (n0 + lr)];
    bm.y = Vb[(size_t)(k0 + 2 * half + 1) * VD_ + (n0 + lr)];
    acc = wmma_f32_4(a, bm, acc);
  }
  float* outb = Out + (size_t)b * Q_ * VD_;
#pragma unroll
  for (int r = 0; r < 8; ++r)
    outb[(size_t)(m0 + r + 8 * half) * VD_ + (n0 + lr)] = acc[r];
}

// ---------------------------------------------------------------------------
extern "C" void kernel_launch(void* const* d_in, const int* in_sizes, int n_in,
                              void* d_out, int out_size, void* d_ws,
                              size_t ws_size, hipStream_t stream) {
  const float* query = (const float*)d_in[0];
  const float* key   = (const float*)d_in[1];
  const float* value = (const float*)d_in[2];
  const float* Wq    = (const float*)d_in[3];
  const float* bq    = (const float*)d_in[4];
  const float* Wk    = (const float*)d_in[5];
  const float* v     = (const float*)d_in[6];

  float* out      = (float*)d_out;
  float* av_out   = out;                                  // (B,Q,VD)
  float* attn_out = out + (size_t)B_ * Q_ * VD_;          // (B,Q,K)

  float* WQp = (float*)d_ws;                              // (B,Q,H)  2 MB
  float* UHt = WQp + (size_t)B_ * Q_ * H_;                // (B,H,K)  4 MB

  proj_q_kernel<<<dim3(H_ / 16, Q_ / 16, B_), 32, 0, stream>>>(query, Wq, bq, WQp);
  proj_k_kernel<<<dim3(KK_ / 16, H_ / 16, B_), 32, 0, stream>>>(key, Wk, UHt);
  scores_softmax_kernel<<<dim3(Q_ / QB_, B_), 256, 0, stream>>>(WQp, UHt, v, attn_out);
  av_kernel<<<dim3(VD_ / 16, Q_ / 16, B_), 32, 0, stream>>>(attn_out, value, av_out);
}